// GraphLayer_3298534883925
// MI455X (gfx1250) — compile-verified
//
#include <hip/hip_runtime.h>
#include <math.h>

#define NN      50000
#define EE      800000
#define DIMF    128
#define HEADSF  4
#define CHF     32
#define NEG_SLOPE 0.2f
#define LN_EPS  1e-5f

typedef __attribute__((ext_vector_type(2))) float v2f;
typedef __attribute__((ext_vector_type(8))) float v8f;

// ---------------------------------------------------------------------------
// Workspace layout (floats)
//   hbuf     : N*128   (h = x@W_gat; later reused as sage_out)
//   gatacc   : N*128   (GAT segment-sum accumulator)
//   sageagg  : N*128   (SAGE neighbor-sum accumulator -> mean in place)
//   a_src    : N*4
//   a_dst    : N*4
//   mbuf     : N*4     (segment max, init -inf)
//   denom    : N*4     (segment sum of exp)
//   cnt      : N       (in-degree as float)
// ---------------------------------------------------------------------------
#define OFF_H      0
#define OFF_GAT    (OFF_H    + NN*DIMF)
#define OFF_SAGE   (OFF_GAT  + NN*DIMF)
#define OFF_ASRC   (OFF_SAGE + NN*DIMF)
#define OFF_ADST   (OFF_ASRC + NN*HEADSF)
#define OFF_M      (OFF_ADST + NN*HEADSF)
#define OFF_DEN    (OFF_M    + NN*HEADSF)
#define OFF_CNT    (OFF_DEN  + NN*HEADSF)

__device__ __forceinline__ void atomicMaxF(float* addr, float val) {
    // float max via integer atomics (correct for mixed signs)
    if (val >= 0.0f) atomicMax((int*)addr, __float_as_int(val));
    else             atomicMin((unsigned int*)addr, __float_as_uint(val));
}

// ---------------- init: zero accumulators, mbuf = -inf ----------------------
__global__ void init_ws(float* __restrict__ gatacc, float* __restrict__ sageagg,
                        float* __restrict__ mbuf, float* __restrict__ denom,
                        float* __restrict__ cnt) {
    const int NA = NN*DIMF;               // 6,400,000
    const int NH = NN*HEADSF;             // 200,000
    const int total = NA + NA + NH + NH + NN;
    for (int i = blockIdx.x*blockDim.x + threadIdx.x; i < total;
         i += gridDim.x*blockDim.x) {
        if      (i < NA)              gatacc[i] = 0.0f;
        else if (i < 2*NA)            sageagg[i - NA] = 0.0f;
        else if (i < 2*NA + NH)       mbuf[i - 2*NA] = -INFINITY;
        else if (i < 2*NA + 2*NH)     denom[i - 2*NA - NH] = 0.0f;
        else                          cnt[i - 2*NA - 2*NH] = 0.0f;
    }
}

// ---------------- WMMA f32 GEMM: C[M,128] (+)= A[M,128] @ W[128,128] --------
// 256 threads = 8 waves (wave32); each wave computes one 16x16 tile via
// V_WMMA_F32_16X16X4_F32, looping K=128 in steps of 4. M, N divisible by 16,
// so EXEC is all ones for every WMMA (ISA requirement).
__global__ void __launch_bounds__(256)
wmma_gemm128(const float* __restrict__ A, const float* __restrict__ W,
             float* __restrict__ C, const float* __restrict__ bias,
             int accumulate) {
    const int wave = threadIdx.x >> 5;      // 0..7 -> 16-col tile
    const int lane = threadIdx.x & 31;
    const int half = lane >> 4;             // 0 or 1
    const int l16  = lane & 15;
    const long m0  = (long)blockIdx.x * 16;
    const int  n0  = wave * 16;

    v8f c;
    if (accumulate) {
#pragma unroll
        for (int r = 0; r < 8; ++r)
            c[r] = C[(m0 + r + 8*half) * DIMF + n0 + l16];
    } else {
#pragma unroll
        for (int r = 0; r < 8; ++r) c[r] = 0.0f;
    }

    const float* arow = A + (m0 + l16) * DIMF;   // A row for this lane's M
#pragma unroll 8
    for (int k0 = 0; k0 < DIMF; k0 += 4) {
        v2f a, b;
        const int ak = k0 + 2*half;
        // A 16x4 frag: lanes 0-15 hold K=k0,k0+1; lanes 16-31 hold K=k0+2,k0+3
        a.x = arow[ak];
        a.y = arow[ak + 1];
        // B 4x16 frag (mirror layout over N)
        b.x = W[ak * DIMF + n0 + l16];
        b.y = W[(ak + 1) * DIMF + n0 + l16];
        c = __builtin_amdgcn_wmma_f32_16x16x4_f32(
                /*neg_a=*/false, a, /*neg_b=*/false, b,
                /*c_mod=*/(short)0, c, /*reuse_a=*/false, /*reuse_b=*/false);
    }

    if (bias) {
        const float bv = bias[n0 + l16];
#pragma unroll
        for (int r = 0; r < 8; ++r) c[r] += bv;
    }
#pragma unroll
    for (int r = 0; r < 8; ++r)
        C[(m0 + r + 8*half) * DIMF + n0 + l16] = c[r];
}

// ---------------- attention scores per (node, head) -------------------------
__global__ void att_scores(const float* __restrict__ hbuf,
                           const float* __restrict__ att_src,
                           const float* __restrict__ att_dst,
                           float* __restrict__ a_src, float* __restrict__ a_dst) {
    const int t = blockIdx.x*blockDim.x + threadIdx.x;
    if (t >= NN*HEADSF) return;
    const int n = t >> 2, hh = t & 3;
    const float* hp = hbuf + (long)n*DIMF + hh*CHF;
    const float* as = att_src + hh*CHF;
    const float* ad = att_dst + hh*CHF;
    float s1 = 0.0f, s2 = 0.0f;
#pragma unroll 8
    for (int c = 0; c < CHF; ++c) { s1 += hp[c]*as[c]; s2 += hp[c]*ad[c]; }
    a_src[t] = s1;
    a_dst[t] = s2;
}

__device__ __forceinline__ float lrelu(float e) {
    return e > 0.0f ? e : NEG_SLOPE * e;
}

// ---------------- edge pass 1: segment max ----------------------------------
__global__ void edge_max(const int* __restrict__ ei,
                         const float* __restrict__ a_src,
                         const float* __restrict__ a_dst,
                         float* __restrict__ mbuf) {
    const int t = blockIdx.x*blockDim.x + threadIdx.x;
    if (t >= (EE + NN)*HEADSF) return;
    const int gid = t >> 2, hh = t & 3;
    int s, d;
    if (gid < EE) { s = ei[gid]; d = ei[EE + gid]; } else { s = d = gid - EE; }
    const float e = lrelu(a_src[s*HEADSF + hh] + a_dst[d*HEADSF + hh]);
    atomicMaxF(&mbuf[d*HEADSF + hh], e);
}

// ---------------- edge pass 2: exp-sum + SAGE degree ------------------------
__global__ void edge_expsum(const int* __restrict__ ei,
                            const float* __restrict__ a_src,
                            const float* __restrict__ a_dst,
                            const float* __restrict__ mbuf,
                            float* __restrict__ denom, float* __restrict__ cnt) {
    const int t = blockIdx.x*blockDim.x + threadIdx.x;
    if (t >= (EE + NN)*HEADSF) return;
    const int gid = t >> 2, hh = t & 3;
    int s, d;
    if (gid < EE) { s = ei[gid]; d = ei[EE + gid]; } else { s = d = gid - EE; }
    const float e  = lrelu(a_src[s*HEADSF + hh] + a_dst[d*HEADSF + hh]);
    const float ex = expf(e - mbuf[d*HEADSF + hh]);
    atomicAdd(&denom[d*HEADSF + hh], ex);
    if (gid < EE && hh == 0) atomicAdd(&cnt[d], 1.0f);
}

// ---------------- edge pass 3: weighted scatter (1 block / edge) ------------
__global__ void edge_aggregate(const int* __restrict__ ei,
                               const float* __restrict__ x,
                               const float* __restrict__ hbuf,
                               const float* __restrict__ a_src,
                               const float* __restrict__ a_dst,
                               const float* __restrict__ mbuf,
                               const float* __restrict__ denom,
                               float* __restrict__ gatacc,
                               float* __restrict__ sageagg) {
    const int gid = blockIdx.x;           // edge (or self-loop) index
    const int c   = threadIdx.x;          // channel 0..127
    const int hh  = c >> 5;               // head
    int s, d;
    if (gid < EE) { s = ei[gid]; d = ei[EE + gid]; } else { s = d = gid - EE; }
    const float e     = lrelu(a_src[s*HEADSF + hh] + a_dst[d*HEADSF + hh]);
    const float alpha = expf(e - mbuf[d*HEADSF + hh]) /
                        (denom[d*HEADSF + hh] + 1e-16f);
    atomicAdd(&gatacc[(long)d*DIMF + c], hbuf[(long)s*DIMF + c] * alpha);
    if (gid < EE)
        atomicAdd(&sageagg[(long)d*DIMF + c], x[(long)s*DIMF + c]);
}

// ---------------- node finalize: SAGE mean in place, gatacc += b_gat --------
__global__ void node_finalize(float* __restrict__ sageagg,
                              float* __restrict__ gatacc,
                              const float* __restrict__ cnt,
                              const float* __restrict__ b_gat) {
    const int t = blockIdx.x*blockDim.x + threadIdx.x;
    if (t >= NN*DIMF) return;
    const int n = t >> 7, c = t & 127;
    float ct = cnt[n];
    ct = ct < 1.0f ? 1.0f : ct;
    sageagg[t] = sageagg[t] / ct;
    gatacc[t]  = gatacc[t] + b_gat[c];
}

// ---------------- residual + LayerNorm (1 block / node row) ----------------
__global__ void ln_kernel(float* __restrict__ out, const float* __restrict__ x,
                          const float* __restrict__ gamma,
                          const float* __restrict__ beta) {
    __shared__ float red[DIMF];
    const int n = blockIdx.x, c = threadIdx.x;
    const float v = out[(long)n*DIMF + c] + x[(long)n*DIMF + c];  // residual
    red[c] = v;
    __syncthreads();
    for (int s = 64; s > 0; s >>= 1) { if (c < s) red[c] += red[c + s]; __syncthreads(); }
    const float mu = red[0] * (1.0f/DIMF);
    __syncthreads();
    const float dv = v - mu;
    red[c] = dv * dv;
    __syncthreads();
    for (int s = 64; s > 0; s >>= 1) { if (c < s) red[c] += red[c + s]; __syncthreads(); }
    const float var = red[0] * (1.0f/DIMF);
    out[(long)n*DIMF + c] = dv * rsqrtf(var + LN_EPS) * gamma[c] + beta[c];
}

// ---------------------------------------------------------------------------
extern "C" void kernel_launch(void* const* d_in, const int* in_sizes, int n_in,
                              void* d_out, int out_size, void* d_ws, size_t ws_size,
                              hipStream_t stream) {
    const float* x        = (const float*)d_in[0];
    const int*   ei       = (const int*)  d_in[1];
    const float* W_gat    = (const float*)d_in[2];
    const float* att_src  = (const float*)d_in[3];
    const float* att_dst  = (const float*)d_in[4];
    const float* b_gat    = (const float*)d_in[5];
    const float* W_sage_l = (const float*)d_in[6];
    const float* b_sage_l = (const float*)d_in[7];
    const float* W_sage_r = (const float*)d_in[8];
    const float* W_proj   = (const float*)d_in[9];
    const float* b_proj   = (const float*)d_in[10];
    const float* gamma    = (const float*)d_in[11];
    const float* beta     = (const float*)d_in[12];
    float* out = (float*)d_out;

    float* ws      = (float*)d_ws;
    float* hbuf    = ws + OFF_H;       // later reused as sage_out
    float* gatacc  = ws + OFF_GAT;
    float* sageagg = ws + OFF_SAGE;
    float* a_src   = ws + OFF_ASRC;
    float* a_dst   = ws + OFF_ADST;
    float* mbuf    = ws + OFF_M;
    float* denom   = ws + OFF_DEN;
    float* cnt     = ws + OFF_CNT;

    const int mtiles = NN / 16;                         // 3125
    const int eh     = (EE + NN) * HEADSF;              // 3,400,000

    // 1. init accumulators
    init_ws<<<2048, 256, 0, stream>>>(gatacc, sageagg, mbuf, denom, cnt);
    // 2. h = x @ W_gat
    wmma_gemm128<<<mtiles, 256, 0, stream>>>(x, W_gat, hbuf, nullptr, 0);
    // 3. attention scores
    att_scores<<<(NN*HEADSF + 255)/256, 256, 0, stream>>>(hbuf, att_src, att_dst,
                                                          a_src, a_dst);
    // 4-6. edge passes (E real edges + N self loops for GAT)
    edge_max   <<<(eh + 255)/256, 256, 0, stream>>>(ei, a_src, a_dst, mbuf);
    edge_expsum<<<(eh + 255)/256, 256, 0, stream>>>(ei, a_src, a_dst, mbuf,
                                                    denom, cnt);
    edge_aggregate<<<EE + NN, DIMF, 0, stream>>>(ei, x, hbuf, a_src, a_dst,
                                                 mbuf, denom, gatacc, sageagg);
    // 7. SAGE mean + b_gat fold
    node_finalize<<<(NN*DIMF + 255)/256, 256, 0, stream>>>(sageagg, gatacc,
                                                           cnt, b_gat);
    // 8-9. sage_out (reuse hbuf) = mean @ W_sage_l + b_sage_l + x @ W_sage_r
    wmma_gemm128<<<mtiles, 256, 0, stream>>>(sageagg, W_sage_l, hbuf, b_sage_l, 0);
    wmma_gemm128<<<mtiles, 256, 0, stream>>>(x, W_sage_r, hbuf, nullptr, 1);
    // 10-11. out = gat_out @ Wp[0:128] + b_proj + sage_out @ Wp[128:256]
    wmma_gemm128<<<mtiles, 256, 0, stream>>>(gatacc, W_proj, out, b_proj, 0);
    wmma_gemm128<<<mtiles, 256, 0, stream>>>(hbuf, W_proj + DIMF*DIMF, out,
                                             nullptr, 1);
    // 12. residual + LayerNorm
    ln_kernel<<<NN, DIMF, 0, stream>>>(out, x, gamma, beta);
}